// GCNModel_7997229105214
// MI455X (gfx1250) — compile-verified
//
#include <hip/hip_runtime.h>
#include <hip/hip_bf16.h>

#define N_USER 100000
#define N_ITEM 200000
#define NTOT   300000   // N_USER + N_ITEM
#define NHID   128
#define E_UI   2000000
#define E_UU   800000
#define NEG_SLOPE 0.5f

typedef __attribute__((ext_vector_type(2))) float v2f;
typedef __attribute__((ext_vector_type(8))) float v8f;

// ---------------------------------------------------------------- utilities
__global__ void zero_f4_kernel(float* __restrict__ p, long long n4) {
    long long i = (long long)blockIdx.x * blockDim.x + threadIdx.x;
    if (i < n4) ((float4*)p)[i] = make_float4(0.f, 0.f, 0.f, 0.f);
}

__global__ void zero_i_kernel(int* __restrict__ p, int n) {
    int i = blockIdx.x * blockDim.x + threadIdx.x;
    if (i < n) p[i] = 0;
}

__global__ void copy_f4_kernel(const float* __restrict__ s, float* __restrict__ d, long long n4) {
    long long i = (long long)blockIdx.x * blockDim.x + threadIdx.x;
    if (i < n4) ((float4*)d)[i] = ((const float4*)s)[i];
}

// --------------------------------------------------------------- degree prep
__global__ void count_kernel(const int* __restrict__ idx, int* __restrict__ cnt, int n) {
    int i = blockIdx.x * blockDim.x + threadIdx.x;
    if (i < n) atomicAdd(&cnt[idx[i]], 1);
}

__global__ void deg_rsqrt_kernel(const int* __restrict__ cnt, float* __restrict__ o, int n) {
    int i = blockIdx.x * blockDim.x + threadIdx.x;
    if (i < n) {
        int c = cnt[i];
        if (c < 1) c = 1;
        o[i] = rsqrtf((float)c);   // clip(deg,1)^-0.5
    }
}

// ------------------------------------------------------------------ GEMM
// out[r,:] = (emb[r,:] @ W) * scale[r], W is 128x128.
//
// W is staged into LDS in a WMMA-fragment-packed layout:
//   lW2[kp*128 + (col ^ swz(kp))] = float2{ W[2kp][col], W[2kp+1][col] }
// where swz(kp) = 16 if kp is odd else 0.  A B-fragment for k-row pair kr
// (kr = k0 + 2*hi, always even) is then ONE contiguous ds_load_b64 into an
// aligned VGPR pair — no register marshalling before v_wmma.  The col^16
// swizzle for odd kp makes the two lane-halves (even kp for lanes 0-15, odd
// kp for lanes 16-31) hit disjoint 32-bank sets -> conflict-free b64 reads.
//
// 256 threads = 8 waves; wave w computes rows [blk*128 + w*16, +16) x 128 cols
// via V_WMMA_F32_16X16X4_F32.  nrows is always a multiple of 16 here, so each
// wave tile is fully in or fully out (EXEC stays all-ones around WMMA).
__global__ void __launch_bounds__(256)
gemm128_wmma_kernel(const float* __restrict__ emb,
                    const float* __restrict__ W,
                    const float* __restrict__ scale,
                    float* __restrict__ out,
                    int nrows) {
    __shared__ v2f lW2[64 * 128];                   // 64 KB of the 320 KB WGP LDS
    {
        float* lw = (float*)lW2;
        for (int i4 = threadIdx.x; i4 < 128 * 32; i4 += 256) {
            const int idx = i4 * 4;
            const int r = idx >> 7;
            const int c = idx & 127;                // multiple of 4
            const float4 w = ((const float4*)W)[i4];
            const int kp  = r >> 1;
            const int par = r & 1;
            const int cs  = c ^ ((kp & 1) << 4);    // bit4 swizzle; c%4==0 so +j safe
            const int b0  = (kp * 128 + cs) * 2 + par;
            lw[b0 + 0] = w.x;
            lw[b0 + 2] = w.y;
            lw[b0 + 4] = w.z;
            lw[b0 + 6] = w.w;
        }
    }
    __syncthreads();

    const int wave = threadIdx.x >> 5;
    const int lane = threadIdx.x & 31;
    const int row0 = (blockIdx.x * 8 + wave) * 16;
    if (row0 >= nrows) return;                      // wave-uniform exit

    const int lo = lane & 15;                       // 0..15
    const int hi = lane >> 4;                       // 0 or 1
    const int sw = hi << 4;                         // odd-kp column swizzle

    v8f acc[8];
#pragma unroll
    for (int nt = 0; nt < 8; ++nt) acc[nt] = (v8f)0.0f;

    for (int k0 = 0; k0 < 128; k0 += 4) {
        // A frag (16x4 f32): lanes 0-15 hold K=k0,k0+1; lanes 16-31 hold K=k0+2,k0+3
        const float* ap = emb + (size_t)(row0 + lo) * NHID + k0 + 2 * hi;
        v2f a; a.x = ap[0]; a.y = ap[1];
        const int kp = (k0 >> 1) + hi;              // this lane-half's k-row pair
#pragma unroll
        for (int nt = 0; nt < 8; ++nt) {
            const v2f b = lW2[kp * 128 + ((nt * 16 + lo) ^ sw)];
            acc[nt] = __builtin_amdgcn_wmma_f32_16x16x4_f32(
                false, a, false, b, (short)0, acc[nt], false, false);
        }
    }

    // D layout: VGPR j -> row j (lanes 0-15) / row j+8 (lanes 16-31), col = tile+lo
#pragma unroll
    for (int j = 0; j < 8; ++j) {
        const int row = row0 + j + hi * 8;
        const float s = scale[row];
#pragma unroll
        for (int nt = 0; nt < 8; ++nt)
            out[(size_t)row * NHID + nt * 16 + lo] = acc[nt][j] * s;
    }
}

// ------------------------------------------------------------------ SpMM
// agg[dst[e], :] += node_f[src[e], :]  (copy_u / sum).
// 32 lanes per edge, float4 per lane; edge indices are wave-uniform -> scalar
// loads, and the 128B row segments are fully coalesced (node_f lives in L2:
// 300000*128*4B = 153.6 MB < 192 MB global L2).
__global__ void spmm_scatter_kernel(const float* __restrict__ node_f,
                                    const int* __restrict__ src,
                                    const int* __restrict__ dst,
                                    float* __restrict__ agg,
                                    int nedges) {
    long long t = (long long)blockIdx.x * blockDim.x + threadIdx.x;
    int e = (int)(t >> 5);
    if (e >= nedges) return;
    const int c4 = ((int)t & 31) * 4;
    const int s = src[e];
    const int d = dst[e];
    const float4 v = *(const float4*)(node_f + (size_t)s * NHID + c4);
    float* p = agg + (size_t)d * NHID + c4;
    atomicAdd(p + 0, v.x);
    atomicAdd(p + 1, v.y);
    atomicAdd(p + 2, v.z);
    atomicAdd(p + 3, v.w);
}

// ------------------------------------------------------------------ post
// emb = leaky_relu(agg * indeg^-0.5) (in place in xbuf), all_emb += l2norm(emb)
// One wave per 128-float row; shuffle reduction for the row norm (wave32).
__global__ void post_kernel(float* __restrict__ xbuf,
                            const float* __restrict__ in_rs,
                            float* __restrict__ all_emb,
                            int nrows) {
    const int wave = threadIdx.x >> 5;
    const int lane = threadIdx.x & 31;
    const int row = blockIdx.x * 8 + wave;
    if (row >= nrows) return;
    const float s = in_rs[row];
    float4 v = ((const float4*)xbuf)[(size_t)row * 32 + lane];
    v.x *= s; v.y *= s; v.z *= s; v.w *= s;
    v.x = v.x >= 0.f ? v.x : NEG_SLOPE * v.x;
    v.y = v.y >= 0.f ? v.y : NEG_SLOPE * v.y;
    v.z = v.z >= 0.f ? v.z : NEG_SLOPE * v.z;
    v.w = v.w >= 0.f ? v.w : NEG_SLOPE * v.w;

    float ss = v.x * v.x + v.y * v.y + v.z * v.z + v.w * v.w;
#pragma unroll
    for (int m = 16; m >= 1; m >>= 1) ss += __shfl_xor(ss, m, 32);
    const float inv = 1.0f / fmaxf(sqrtf(ss), 1e-12f);

    float4* ap = (float4*)all_emb + (size_t)row * 32 + lane;
    float4 a = *ap;
    a.x += v.x * inv; a.y += v.y * inv; a.z += v.z * inv; a.w += v.w * inv;
    *ap = a;

    ((float4*)xbuf)[(size_t)row * 32 + lane] = v;   // emb for next layer
}

// ------------------------------------------------------------------ driver
static inline unsigned cdiv(long long a, long long b) { return (unsigned)((a + b - 1) / b); }

extern "C" void kernel_launch(void* const* d_in, const int* in_sizes, int n_in,
                              void* d_out, int out_size, void* d_ws, size_t ws_size,
                              hipStream_t stream) {
    const float* user_emb = (const float*)d_in[0];   // [100000,128]
    const float* item_emb = (const float*)d_in[1];   // [200000,128]
    const float* ui_u_w   = (const float*)d_in[2];   // [3,128,128]
    const float* ui_v_w   = (const float*)d_in[3];   // [3,128,128]
    const float* uu_u_w   = (const float*)d_in[4];   // [2,128,128]
    const int*   src_ui   = (const int*)d_in[5];
    const int*   dst_ui   = (const int*)d_in[6];
    const int*   src_uu   = (const int*)d_in[7];
    const int*   dst_uu   = (const int*)d_in[8];

    float* out_ui = (float*)d_out;                        // [300000,128]
    float* out_uu = out_ui + (size_t)NTOT * NHID;         // [100000,128]

    // workspace layout (floats):  X | Y | degree tables | int counts
    float* X         = (float*)d_ws;                      // emb / scatter target
    float* Y         = X + (size_t)NTOT * NHID;           // node_f
    float* ui_out_rs = Y + (size_t)NTOT * NHID;
    float* ui_in_rs  = ui_out_rs + NTOT;
    float* uu_out_rs = ui_in_rs + NTOT;
    float* uu_in_rs  = uu_out_rs + N_USER;
    int*   cnt_a     = (int*)(uu_in_rs + N_USER);
    int*   cnt_b     = cnt_a + NTOT;

    const long long user4 = (long long)N_USER * NHID / 4;
    const long long item4 = (long long)N_ITEM * NHID / 4;
    const long long tot4  = (long long)NTOT * NHID / 4;

    // ---- UI degree tables (loop invariant)
    zero_i_kernel<<<cdiv(NTOT, 256), 256, 0, stream>>>(cnt_a, NTOT);
    zero_i_kernel<<<cdiv(NTOT, 256), 256, 0, stream>>>(cnt_b, NTOT);
    count_kernel<<<cdiv(E_UI, 256), 256, 0, stream>>>(src_ui, cnt_a, E_UI);
    count_kernel<<<cdiv(E_UI, 256), 256, 0, stream>>>(dst_ui, cnt_b, E_UI);
    deg_rsqrt_kernel<<<cdiv(NTOT, 256), 256, 0, stream>>>(cnt_a, ui_out_rs, NTOT);
    deg_rsqrt_kernel<<<cdiv(NTOT, 256), 256, 0, stream>>>(cnt_b, ui_in_rs, NTOT);

    // ---- init: X = concat(user,item); out_ui = same (all_emb starts at init_emb)
    copy_f4_kernel<<<cdiv(user4, 256), 256, 0, stream>>>(user_emb, X, user4);
    copy_f4_kernel<<<cdiv(item4, 256), 256, 0, stream>>>(item_emb, X + (size_t)N_USER * NHID, item4);
    copy_f4_kernel<<<cdiv(user4, 256), 256, 0, stream>>>(user_emb, out_ui, user4);
    copy_f4_kernel<<<cdiv(item4, 256), 256, 0, stream>>>(item_emb, out_ui + (size_t)N_USER * NHID, item4);

    // ---- UI propagation: 3 layers
    for (int l = 0; l < 3; ++l) {
        gemm128_wmma_kernel<<<cdiv(N_USER, 128), 256, 0, stream>>>(
            X, ui_u_w + (size_t)l * NHID * NHID, ui_out_rs, Y, N_USER);
        gemm128_wmma_kernel<<<cdiv(N_ITEM, 128), 256, 0, stream>>>(
            X + (size_t)N_USER * NHID, ui_v_w + (size_t)l * NHID * NHID,
            ui_out_rs + N_USER, Y + (size_t)N_USER * NHID, N_ITEM);
        zero_f4_kernel<<<cdiv(tot4, 256), 256, 0, stream>>>(X, tot4);
        spmm_scatter_kernel<<<cdiv((long long)E_UI * 32, 256), 256, 0, stream>>>(
            Y, src_ui, dst_ui, X, E_UI);
        post_kernel<<<cdiv(NTOT, 8), 256, 0, stream>>>(X, ui_in_rs, out_ui, NTOT);
    }

    // ---- UU degree tables
    zero_i_kernel<<<cdiv(N_USER, 256), 256, 0, stream>>>(cnt_a, N_USER);
    zero_i_kernel<<<cdiv(N_USER, 256), 256, 0, stream>>>(cnt_b, N_USER);
    count_kernel<<<cdiv(E_UU, 256), 256, 0, stream>>>(src_uu, cnt_a, E_UU);
    count_kernel<<<cdiv(E_UU, 256), 256, 0, stream>>>(dst_uu, cnt_b, E_UU);
    deg_rsqrt_kernel<<<cdiv(N_USER, 256), 256, 0, stream>>>(cnt_a, uu_out_rs, N_USER);
    deg_rsqrt_kernel<<<cdiv(N_USER, 256), 256, 0, stream>>>(cnt_b, uu_in_rs, N_USER);

    // ---- init UU: X = user_emb; out_uu = user_emb
    copy_f4_kernel<<<cdiv(user4, 256), 256, 0, stream>>>(user_emb, X, user4);
    copy_f4_kernel<<<cdiv(user4, 256), 256, 0, stream>>>(user_emb, out_uu, user4);

    // ---- UU propagation: 2 layers
    for (int l = 0; l < 2; ++l) {
        gemm128_wmma_kernel<<<cdiv(N_USER, 128), 256, 0, stream>>>(
            X, uu_u_w + (size_t)l * NHID * NHID, uu_out_rs, Y, N_USER);
        zero_f4_kernel<<<cdiv(user4, 256), 256, 0, stream>>>(X, user4);
        spmm_scatter_kernel<<<cdiv((long long)E_UU * 32, 256), 256, 0, stream>>>(
            Y, src_uu, dst_uu, X, E_UU);
        post_kernel<<<cdiv(N_USER, 8), 256, 0, stream>>>(X, uu_in_rs, out_uu, N_USER);
    }
}